// LSTM_71133248356703
// MI455X (gfx1250) — compile-verified
//
#include <hip/hip_runtime.h>
#include <math.h>

// ---------------------------------------------------------------------------
// Tiny 3-layer LSTM policy inference (batch=1) for MI455X / gfx1250.
//
// Problem is launch-latency bound (~5 KFLOP, ~11 KB weights -> ~0.5 ns of HBM
// time at 23.3 TB/s), so: ONE kernel, ONE workgroup, ONE wave32. All four
// chained mat-vecs are mapped onto V_WMMA_F32_16X16X4_F32:
//   D = A x B,  A[m][k] = x[k] (broadcast down rows, documented A layout),
//               B[k][n] = W[n][k] (two 16-wide N tiles -> 32 gates),
//   => every row of D is the gate vector; lane l reads gates[l] directly
//      from accumulator VGPR 0 per the documented C/D layout.
// Reference quirk: h0/c0 are re-zeroed each layer, so w_hh* never matters and
//   c_new = sigmoid(i)*tanh(g),  h_new = sigmoid(o)*tanh(c_new).
// ---------------------------------------------------------------------------

typedef float v2f __attribute__((ext_vector_type(2)));
typedef float v8f __attribute__((ext_vector_type(8)));

#define OBS_DIM 64
#define HIDDEN  8
#define GATES   32   // 4*HIDDEN

__device__ __forceinline__ v8f wmma4(v2f a, v2f b, v8f c) {
  // (neg_a, A, neg_b, B, c_mod, C, reuse_a, reuse_b)
  return __builtin_amdgcn_wmma_f32_16x16x4_f32(false, a, false, b,
                                               (short)0, c, false, false);
}

__device__ __forceinline__ float sigmoid_f(float x) {
  return 1.0f / (1.0f + expf(-x));
}

// Post-WMMA: bias add, gate nonlinearities (one gate per lane), combine
// i/g/o through LDS into the 8-wide hidden state.
__device__ __forceinline__ void gates_to_h(v8f acc0, v8f acc1,
                                           const float* __restrict__ b_ih,
                                           const float* __restrict__ b_hh,
                                           float* g_sh, float* h_sh, int lane) {
  // lane l holds gates[l]: lanes 0-15 from tile0 (N 0..15), 16-31 from tile1.
  float g = ((lane < 16) ? acc0[0] : acc1[0]) + b_ih[lane] + b_hh[lane];
  // gate groups: i=0..7, f=8..15, g=16..23, o=24..31
  float act = ((lane >> 3) == 2) ? tanhf(g) : sigmoid_f(g);
  g_sh[lane] = act;
  __syncthreads();                       // single wave: ~free
  if (lane < HIDDEN) {
    float iv = g_sh[lane];
    float gv = g_sh[lane + 16];
    float ov = g_sh[lane + 24];
    h_sh[lane] = ov * tanhf(iv * gv);    // f*c0 == 0
  }
  __syncthreads();
}

__global__ __launch_bounds__(32) void lstm_policy_kernel(
    const float* __restrict__ obs,
    const float* __restrict__ w_ih0, const float* __restrict__ b_ih0,
    const float* __restrict__ b_hh0,
    const float* __restrict__ w_ih1, const float* __restrict__ b_ih1,
    const float* __restrict__ b_hh1,
    const float* __restrict__ w_ih2, const float* __restrict__ b_ih2,
    const float* __restrict__ b_hh2,
    const float* __restrict__ w_out, const float* __restrict__ b_out,
    float* __restrict__ out) {
  const int lane = threadIdx.x;    // 0..31 == full wave32; EXEC all ones
  const int half = lane >> 4;      // selects K sub-pair in A/B operand layout
  const int n    = lane & 15;      // column within a 16-wide N tile

  __shared__ float g_sh[GATES];
  __shared__ float h_sh[HIDDEN];

  const v8f vzero = {0.f, 0.f, 0.f, 0.f, 0.f, 0.f, 0.f, 0.f};

  // ----- layer 0: gates = w_ih0 (32x64) * obs, K = 64 -> 16 K-chunks -----
  {
    v8f acc0 = vzero, acc1 = vzero;               // two independent chains
    const float* wa = w_ih0 + (size_t)n * OBS_DIM;         // rows n
    const float* wb = w_ih0 + (size_t)(16 + n) * OBS_DIM;  // rows 16+n
#pragma unroll
    for (int kc = 0; kc < OBS_DIM / 4; ++kc) {
      const int k0 = kc * 4 + 2 * half;
      v2f a; a.x = obs[k0]; a.y = obs[k0 + 1];    // A[m][k] = x[k] (bcast)
      v2f b0 = *(const v2f*)(wa + k0);            // B[k][n] = W[n][k]
      v2f b1 = *(const v2f*)(wb + k0);
      acc0 = wmma4(a, b0, acc0);
      acc1 = wmma4(a, b1, acc1);
    }
    gates_to_h(acc0, acc1, b_ih0, b_hh0, g_sh, h_sh, lane);
  }

  // ----- layers 1 & 2: gates = w_ih (32x8) * h, K = 8 -> 2 K-chunks -----
  const float* wl[2]  = {w_ih1, w_ih2};
  const float* bil[2] = {b_ih1, b_ih2};
  const float* bhl[2] = {b_hh1, b_hh2};
#pragma unroll
  for (int layer = 0; layer < 2; ++layer) {
    v8f acc0 = vzero, acc1 = vzero;
    const float* wa = wl[layer] + (size_t)n * HIDDEN;
    const float* wb = wl[layer] + (size_t)(16 + n) * HIDDEN;
#pragma unroll
    for (int kc = 0; kc < 2; ++kc) {
      const int k0 = kc * 4 + 2 * half;
      v2f a; a.x = h_sh[k0]; a.y = h_sh[k0 + 1];
      v2f b0 = *(const v2f*)(wa + k0);
      v2f b1 = *(const v2f*)(wb + k0);
      acc0 = wmma4(a, b0, acc0);
      acc1 = wmma4(a, b1, acc1);
    }
    gates_to_h(acc0, acc1, bil[layer], bhl[layer], g_sh, h_sh, lane);
  }

  // ----- head: action = sigmoid(w_out (8x8) * h + b_out), one N tile -----
  {
    v8f acc = vzero;
#pragma unroll
    for (int kc = 0; kc < 2; ++kc) {
      const int k0 = kc * 4 + 2 * half;
      v2f a; a.x = h_sh[k0]; a.y = h_sh[k0 + 1];
      v2f b;
      if (n < 8) {                       // valid output rows 0..7
        b = *(const v2f*)(w_out + (size_t)n * HIDDEN + k0);
      } else {
        b.x = 0.f; b.y = 0.f;            // pad columns 8..15 with zeros
      }
      acc = wmma4(a, b, acc);            // EXEC all ones here (no divergence)
    }
    if (lane < 8) {
      out[lane] = sigmoid_f(acc[0] + b_out[lane]);
    }
  }
}

extern "C" void kernel_launch(void* const* d_in, const int* in_sizes, int n_in,
                              void* d_out, int out_size, void* d_ws,
                              size_t ws_size, hipStream_t stream) {
  (void)in_sizes; (void)n_in; (void)out_size; (void)d_ws; (void)ws_size;
  const float* obs   = (const float*)d_in[0];
  const float* w_ih0 = (const float*)d_in[1];
  // d_in[2] = w_hh0: mathematically unused (h0 == 0 in reference)
  const float* b_ih0 = (const float*)d_in[3];
  const float* b_hh0 = (const float*)d_in[4];
  const float* w_ih1 = (const float*)d_in[5];
  const float* b_ih1 = (const float*)d_in[7];
  const float* b_hh1 = (const float*)d_in[8];
  const float* w_ih2 = (const float*)d_in[9];
  const float* b_ih2 = (const float*)d_in[11];
  const float* b_hh2 = (const float*)d_in[12];
  const float* w_out = (const float*)d_in[13];
  const float* b_out = (const float*)d_in[14];

  lstm_policy_kernel<<<1, 32, 0, stream>>>(
      obs, w_ih0, b_ih0, b_hh0, w_ih1, b_ih1, b_hh1, w_ih2, b_ih2, b_hh2,
      w_out, b_out, (float*)d_out);
}